// RSSM_48730698940624
// MI455X (gfx1250) — compile-verified
//
#include <hip/hip_runtime.h>
#include <math.h>

typedef float v2f __attribute__((ext_vector_type(2)));
typedef float v8f __attribute__((ext_vector_type(8)));

#define WMMA4(a, b, c) \
  __builtin_amdgcn_wmma_f32_16x16x4_f32(false, (a), false, (b), (short)0, (c), false, false)

static constexpr int kB = 512, kT = 64, kE = 1024, kA = 32, kS = 32, kD = 512, kH = 512;
static constexpr int OUTW = 2 * (3 * kS + kD);  // 1216 floats per (b,t)
static constexpr int AST  = 516;                // padded LDS row stride (516 % 64 == 4 -> conflict-free)
static constexpr int SST  = 132;                // stats row stride (132 % 64 == 4)

__device__ __forceinline__ float dsigm(float x) { return 1.f / (1.f + expf(-x)); }
__device__ __forceinline__ float delu(float x)  { return x > 0.f ? x : expm1f(x); }
__device__ __forceinline__ float dsoftplus(float x) {
  return fmaxf(x, 0.f) + log1pf(expf(-fabsf(x)));
}

// ---------------------------------------------------------------------------
// Precompute 1: ac[t*B+b][h] = action[b,t,:] @ ini0_w[h,32:64] + ini0_b[h]
// ---------------------------------------------------------------------------
__global__ void rssm_ac(const float* __restrict__ action,
                        const float* __restrict__ ini0_w,
                        const float* __restrict__ ini0_b,
                        float* __restrict__ ac) {
  int o = blockIdx.x * 256 + threadIdx.x;
  if (o >= kT * kB * kH) return;
  int h = o & (kH - 1);
  int r = o >> 9;            // r = t*B + b
  int t = r >> 9;
  int b = r & (kB - 1);
  const float* arow = action + ((size_t)b * kT + t) * kA;
  const float* wrow = ini0_w + (size_t)h * (kS + kA) + kS;
  float s = ini0_b[h];
  #pragma unroll
  for (int j = 0; j < kA; ++j) s += arow[j] * wrow[j];
  ac[o] = s;
}

// ---------------------------------------------------------------------------
// Precompute 2 (WMMA): ec[t*B+b][n] = embed[b,t,:] @ obi0_w[n,512:1536] + obi0_b[n]
// M = T*B = 32768, N = 512, K = 1024.  Each WG: 16 rows x 128 cols, 8 waves,
// one 16x16 tile per wave, K staged through LDS in 128-wide chunks.
// ---------------------------------------------------------------------------
__global__ __launch_bounds__(256, 1)
void rssm_ec(const float* __restrict__ embed,
             const float* __restrict__ obi0_w,
             const float* __restrict__ obi0_b,
             float* __restrict__ ec) {
  __shared__ float ebuf[16 * 132];  // 132 % 64 == 4 -> conflict-free fragments

  const int r0   = (blockIdx.x >> 2) * 16;       // row block (t*B+b space)
  const int nb   = (blockIdx.x & 3) * 128;       // col block
  const int tid  = threadIdx.x;
  const int wave = tid >> 5;
  const int lane = tid & 31;
  const int lh   = lane & 15;
  const int khf  = (lane >> 4) << 1;             // 0 or 2
  const int mo   = (lane >> 4) << 3;             // 0 or 8
  const int j0   = nb + wave * 16;

  float bv = obi0_b[j0 + lh];
  v8f acc = {bv, bv, bv, bv, bv, bv, bv, bv};

  for (int k0 = 0; k0 < kE; k0 += 128) {
    __syncthreads();
    // cooperative coalesced load: 16 rows x 128 cols as float4
    for (int i4 = tid; i4 < 16 * 32; i4 += 256) {
      int row = i4 >> 5;
      int c4  = i4 & 31;
      int r   = r0 + row;
      int t   = r >> 9;
      int b   = r & (kB - 1);
      const float4 src =
          *(const float4*)(embed + ((size_t)b * kT + t) * kE + k0 + c4 * 4);
      *(float4*)(&ebuf[row * 132 + c4 * 4]) = src;
    }
    __syncthreads();
    const float* arow = &ebuf[lh * 132 + khf];
    const float* brow = obi0_w + (size_t)(j0 + lh) * (kD + kE) + kD + k0 + khf;
    #pragma unroll 8
    for (int kk = 0; kk < 128; kk += 4)
      acc = WMMA4(*(const v2f*)(arow + kk), *(const v2f*)(brow + kk), acc);
  }
  #pragma unroll
  for (int v = 0; v < 8; ++v)
    ec[(size_t)(r0 + mo + v) * kH + j0 + lh] = acc[v];
}

// ---------------------------------------------------------------------------
// Persistent scan kernel: 32 WGs x 256 threads (8 wave32). WG owns 16 batch
// rows, loops t = 0..63 with recurrent state (stoch, deter) in LDS.
// Phase order per step:
//   A: x = elu(stoch@Wsa^T + ac[t])            -> act   (all waves)
//   B: GRU(x, deter) -> deter_new (in place)   -> det   (all waves)
//   C: y  = elu(deter@imo0^T + b)              -> ybuf  (all waves)
//   F: x2 = elu(deter@obi0_d^T + ec[t])        -> act   (all waves, no barrier vs C)
//   D||G: ims(y) -> stat[:, 0:64]   (waves 0-3)
//         obs(x2)-> stat[:,64:128]  (waves 4-7) concurrently
//   E: merged epilogue: prior/post outputs, deter copies, stoch carry
// ---------------------------------------------------------------------------
__global__ __launch_bounds__(256, 1)
void rssm_scan(const float* __restrict__ ini0_w,
               const float* __restrict__ w_ih, const float* __restrict__ w_hh,
               const float* __restrict__ b_ih, const float* __restrict__ b_hh,
               const float* __restrict__ imo0_w, const float* __restrict__ imo0_b,
               const float* __restrict__ ims_w, const float* __restrict__ ims_b,
               const float* __restrict__ obi0_w,
               const float* __restrict__ obs_w, const float* __restrict__ obs_b,
               const float* __restrict__ ac, const float* __restrict__ ec,
               const float* __restrict__ noise_p, const float* __restrict__ noise_o,
               float* __restrict__ out) {
  __shared__ float act[16 * AST];   // x -> x2 (512 wide, padded)
  __shared__ float det[16 * AST];   // deter (updated in place each step)
  __shared__ float ybuf[16 * AST];  // y = elu(imo0 output)
  __shared__ float stc[16 * 36];    // stoch carry (32 wide, padded)
  __shared__ float stat[16 * SST];  // [prior mean|std_raw | post mean|std_raw]

  const int m0   = blockIdx.x * 16;
  const int tid  = threadIdx.x;
  const int wave = tid >> 5;
  const int lane = tid & 31;
  const int lh   = lane & 15;
  const int khf  = (lane >> 4) << 1;  // K offset 0 / 2 (ISA 16x4 fp32 layout)
  const int mo   = (lane >> 4) << 3;  // C/D row offset 0 / 8

  for (int i = tid; i < 16 * AST; i += 256) { act[i] = 0.f; det[i] = 0.f; }
  for (int i = tid; i < 16 * 36; i += 256) stc[i] = 0.f;
  __syncthreads();

  for (int t = 0; t < kT; ++t) {
    //=================== A: x = elu(stoch @ Wsa^T + ac[t]) ===================
    {
      const float* arow = &stc[lh * 36 + khf];
      #pragma unroll
      for (int tt = 0; tt < 4; ++tt) {
        const int j0 = wave * 64 + tt * 16;
        v8f acc;
        #pragma unroll
        for (int v = 0; v < 8; ++v)
          acc[v] = ac[((size_t)t * kB + m0 + mo + v) * kH + j0 + lh];
        const float* brow = ini0_w + (size_t)(j0 + lh) * (kS + kA) + khf;
        #pragma unroll
        for (int k = 0; k < kS; k += 4)
          acc = WMMA4(*(const v2f*)(arow + k), *(const v2f*)(brow + k), acc);
        #pragma unroll
        for (int v = 0; v < 8; ++v)
          act[(mo + v) * AST + j0 + lh] = delu(acc[v]);
      }
    }
    __syncthreads();

    //=================== B: GRU -> deter_new (in-place) ======================
    float dn[4][8];
    {
      const float* xrow = &act[lh * AST + khf];
      const float* hrow = &det[lh * AST + khf];
      #pragma unroll
      for (int tt = 0; tt < 4; ++tt) {
        const int j0  = wave * 64 + tt * 16;
        const int col = j0 + lh;
        float brv = b_ih[col]        + b_hh[col];
        float bzv = b_ih[kD + col]   + b_hh[kD + col];
        float bnv = b_ih[2*kD + col];
        float bhv = b_hh[2*kD + col];
        v8f ar = {brv, brv, brv, brv, brv, brv, brv, brv};
        v8f az = {bzv, bzv, bzv, bzv, bzv, bzv, bzv, bzv};
        v8f an = {bnv, bnv, bnv, bnv, bnv, bnv, bnv, bnv};
        v8f ah = {bhv, bhv, bhv, bhv, bhv, bhv, bhv, bhv};
        const float* wri = w_ih + (size_t)(col)          * kD + khf;
        const float* wzi = w_ih + (size_t)(kD + col)     * kD + khf;
        const float* wni = w_ih + (size_t)(2 * kD + col) * kD + khf;
        const float* wrh = w_hh + (size_t)(col)          * kD + khf;
        const float* wzh = w_hh + (size_t)(kD + col)     * kD + khf;
        const float* wnh = w_hh + (size_t)(2 * kD + col) * kD + khf;
        #pragma unroll 2
        for (int k = 0; k < kD; k += 4) {
          v2f axv = *(const v2f*)(xrow + k);
          v2f ahv = *(const v2f*)(hrow + k);
          ar = WMMA4(axv, *(const v2f*)(wri + k), ar);
          ar = WMMA4(ahv, *(const v2f*)(wrh + k), ar);
          az = WMMA4(axv, *(const v2f*)(wzi + k), az);
          az = WMMA4(ahv, *(const v2f*)(wzh + k), az);
          an = WMMA4(axv, *(const v2f*)(wni + k), an);
          ah = WMMA4(ahv, *(const v2f*)(wnh + k), ah);
        }
        #pragma unroll
        for (int v = 0; v < 8; ++v) {
          float h = det[(mo + v) * AST + col];
          float r = dsigm(ar[v]);
          float z = dsigm(az[v]);
          float n = tanhf(an[v] + r * ah[v]);
          dn[tt][v] = (1.f - z) * n + z * h;
        }
      }
    }
    __syncthreads();  // all waves done reading act/det
    #pragma unroll
    for (int tt = 0; tt < 4; ++tt) {
      const int j0 = wave * 64 + tt * 16;
      #pragma unroll
      for (int v = 0; v < 8; ++v) det[(mo + v) * AST + j0 + lh] = dn[tt][v];
    }
    __syncthreads();  // det now holds deter_new

    //=================== C: y = elu(deter @ imo0^T + b) -> ybuf ==============
    {
      const float* hrow = &det[lh * AST + khf];
      #pragma unroll
      for (int tt = 0; tt < 4; ++tt) {
        const int j0 = wave * 64 + tt * 16;
        float bv = imo0_b[j0 + lh];
        v8f acc = {bv, bv, bv, bv, bv, bv, bv, bv};
        const float* brow = imo0_w + (size_t)(j0 + lh) * kD + khf;
        #pragma unroll 4
        for (int k = 0; k < kD; k += 4)
          acc = WMMA4(*(const v2f*)(hrow + k), *(const v2f*)(brow + k), acc);
        #pragma unroll
        for (int v = 0; v < 8; ++v)
          ybuf[(mo + v) * AST + j0 + lh] = delu(acc[v]);
      }

      //================ F: x2 = elu(deter @ obi0_d^T + ec[t]) -> act =========
      // No barrier needed vs C: disjoint writes, both read only det.
      #pragma unroll
      for (int tt = 0; tt < 4; ++tt) {
        const int j0 = wave * 64 + tt * 16;
        v8f acc;
        #pragma unroll
        for (int v = 0; v < 8; ++v)
          acc[v] = ec[((size_t)t * kB + m0 + mo + v) * kH + j0 + lh];
        const float* brow = obi0_w + (size_t)(j0 + lh) * (kD + kE) + khf;
        #pragma unroll 4
        for (int k = 0; k < kD; k += 4)
          acc = WMMA4(*(const v2f*)(hrow + k), *(const v2f*)(brow + k), acc);
        #pragma unroll
        for (int v = 0; v < 8; ++v)
          act[(mo + v) * AST + j0 + lh] = delu(acc[v]);
      }
    }
    __syncthreads();

    //======== D || G: prior stats (waves 0-3) and post stats (waves 4-7) =====
    if (wave < 4) {
      const int j0 = wave * 16;
      const float* yrow = &ybuf[lh * AST + khf];
      float bv = ims_b[j0 + lh];
      v8f acc = {bv, bv, bv, bv, bv, bv, bv, bv};
      const float* brow = ims_w + (size_t)(j0 + lh) * kD + khf;
      #pragma unroll 4
      for (int k = 0; k < kD; k += 4)
        acc = WMMA4(*(const v2f*)(yrow + k), *(const v2f*)(brow + k), acc);
      #pragma unroll
      for (int v = 0; v < 8; ++v) stat[(mo + v) * SST + j0 + lh] = acc[v];
    } else {
      const int j0 = (wave - 4) * 16;
      const float* xrow = &act[lh * AST + khf];
      float bv = obs_b[j0 + lh];
      v8f acc = {bv, bv, bv, bv, bv, bv, bv, bv};
      const float* brow = obs_w + (size_t)(j0 + lh) * kD + khf;
      #pragma unroll 4
      for (int k = 0; k < kD; k += 4)
        acc = WMMA4(*(const v2f*)(xrow + k), *(const v2f*)(brow + k), acc);
      #pragma unroll
      for (int v = 0; v < 8; ++v) stat[(mo + v) * SST + 64 + j0 + lh] = acc[v];
    }
    __syncthreads();

    //=================== E: merged epilogue ==================================
    for (int idx = tid; idx < 512; idx += 256) {
      int m = idx >> 5, s = idx & 31;
      size_t base = ((size_t)(m0 + m) * kT + t) * OUTW;
      size_t nidx = ((size_t)t * kB + m0 + m) * kS + s;
      // prior
      float pm = stat[m * SST + s];
      float ps = dsoftplus(stat[m * SST + 32 + s]) + 0.1f;
      out[base + 608 + s] = pm;
      out[base + 640 + s] = ps;
      out[base + 672 + s] = pm + ps * noise_p[nidx];
      // posterior
      float om = stat[m * SST + 64 + s];
      float os = dsoftplus(stat[m * SST + 96 + s]) + 0.1f;
      float post = om + os * noise_o[nidx];
      out[base + s]      = om;
      out[base + 32 + s] = os;
      out[base + 64 + s] = post;
      stc[m * 36 + s]    = post;
    }
    for (int idx = tid; idx < 16 * kD; idx += 256) {
      int m = idx >> 9, c = idx & (kD - 1);
      float v = det[m * AST + c];
      size_t base = ((size_t)(m0 + m) * kT + t) * OUTW;
      out[base + 96 + c]  = v;
      out[base + 704 + c] = v;
    }
    __syncthreads();
  }
}

// ---------------------------------------------------------------------------
extern "C" void kernel_launch(void* const* d_in, const int* in_sizes, int n_in,
                              void* d_out, int out_size, void* d_ws, size_t ws_size,
                              hipStream_t stream) {
  (void)in_sizes; (void)n_in; (void)out_size; (void)ws_size;
  const float* embed   = (const float*)d_in[0];
  const float* action  = (const float*)d_in[1];
  const float* noise_p = (const float*)d_in[2];
  const float* noise_o = (const float*)d_in[3];
  const float* ini0_w  = (const float*)d_in[4];
  const float* ini0_b  = (const float*)d_in[5];
  const float* w_ih    = (const float*)d_in[6];
  const float* w_hh    = (const float*)d_in[7];
  const float* b_ih    = (const float*)d_in[8];
  const float* b_hh    = (const float*)d_in[9];
  const float* imo0_w  = (const float*)d_in[10];
  const float* imo0_b  = (const float*)d_in[11];
  const float* ims_w   = (const float*)d_in[12];
  const float* ims_b   = (const float*)d_in[13];
  const float* obi0_w  = (const float*)d_in[14];
  const float* obi0_b  = (const float*)d_in[15];
  const float* obs_w   = (const float*)d_in[16];
  const float* obs_b   = (const float*)d_in[17];
  float* out = (float*)d_out;

  float* ac = (float*)d_ws;                       // [T*B][H]
  float* ec = ac + (size_t)kT * kB * kH;          // [T*B][H]

  rssm_ac<<<(kT * kB * kH + 255) / 256, 256, 0, stream>>>(action, ini0_w, ini0_b, ac);
  rssm_ec<<<dim3(2048 * 4), 256, 0, stream>>>(embed, obi0_w, obi0_b, ec);
  rssm_scan<<<dim3(kB / 16), 256, 0, stream>>>(
      ini0_w, w_ih, w_hh, b_ih, b_hh, imo0_w, imo0_b, ims_w, ims_b,
      obi0_w, obs_w, obs_b, ac, ec, noise_p, noise_o, out);
}